// SAGE_34376918237987
// MI455X (gfx1250) — compile-verified
//
#include <hip/hip_runtime.h>

typedef __attribute__((ext_vector_type(2))) float v2f;
typedef __attribute__((ext_vector_type(8))) float v8f;

#define N_NODES 50000
#define N_EDGES 800000
#define FEATS 64

// ---------------------------------------------------------------------------
// Degree accumulation: deg[dst[e]] += 1.0f
// ---------------------------------------------------------------------------
__global__ void sage_deg_kernel(const int* __restrict__ dst, int nE,
                                float* __restrict__ deg) {
    int e = blockIdx.x * blockDim.x + threadIdx.x;
    if (e < nE) atomicAdd(&deg[dst[e]], 1.0f);
}

// deg -> 1/max(deg,1) in place
__global__ void sage_invdeg_kernel(float* __restrict__ deg, int n) {
    int i = blockIdx.x * blockDim.x + threadIdx.x;
    if (i < n) deg[i] = 1.0f / fmaxf(deg[i], 1.0f);
}

// ---------------------------------------------------------------------------
// Edge scatter: agg[dst[e]][:] += X[src[e]][:]
// 16 threads per edge, each owning 4 consecutive feats (float4 gather,
// 4 x global_atomic_add_f32 scatter; no-return atomics ride STOREcnt).
// ---------------------------------------------------------------------------
__global__ void __launch_bounds__(256)
sage_scatter_kernel(const float* __restrict__ X,
                    const int* __restrict__ src, const int* __restrict__ dst,
                    int nE, float* __restrict__ agg) {
    long t = (long)blockIdx.x * blockDim.x + threadIdx.x;
    int e = (int)(t >> 4);
    if (e >= nE) return;
    int c = ((int)t & 15) * 4;
    int s = src[e];
    int d = dst[e];
    const float4 v = *(const float4*)(X + (long)s * FEATS + c);
    float* base = agg + (long)d * FEATS + c;
    atomicAdd(base + 0, v.x);
    atomicAdd(base + 1, v.y);
    atomicAdd(base + 2, v.z);
    atomicAdd(base + 3, v.w);
}

// ---------------------------------------------------------------------------
// Fused SAGE combine via fp32 WMMA:
//   OUT[16x64 tile] = act( X_tile @ Wself + (AGG_tile * inv) @ Wneigh + b )
// One wave per 16-row node tile; 4 N-tiles x 16 K-steps x 2 matrices
// = 128 v_wmma_f32_16x16x4_f32 per tile. Weights staged in LDS (32 KB).
// ---------------------------------------------------------------------------
__global__ void __launch_bounds__(256)
sage_combine_kernel(const float* __restrict__ X,
                    const float* __restrict__ AGG,
                    const float* __restrict__ INV,
                    const float* __restrict__ Wself,
                    const float* __restrict__ Wneigh,
                    const float* __restrict__ bias,
                    float* __restrict__ OUT,
                    int nnodes, int do_relu) {
    __shared__ float sWs[FEATS * FEATS];
    __shared__ float sWn[FEATS * FEATS];
    __shared__ float sB[FEATS];

    for (int i = threadIdx.x; i < FEATS * FEATS; i += 256) {
        sWs[i] = Wself[i];
        sWn[i] = Wneigh[i];
    }
    if (threadIdx.x < FEATS) sB[threadIdx.x] = bias[threadIdx.x];
    __syncthreads();

    const int wave = threadIdx.x >> 5;              // 0..7
    const int lane = threadIdx.x & 31;
    const int tile = blockIdx.x * 8 + wave;
    const int row0 = tile * 16;
    if (row0 >= nnodes) return;                     // whole-wave uniform exit

    const int r  = lane & 15;                       // M (A) / N (B,C)
    const int hi = lane >> 4;                       // K-pair select
    const int row = row0 + r;
    const float inv = INV[row];

    v8f acc[4] = {};                                // 4 N-tiles of 16x16 f32

    #pragma unroll 4
    for (int k = 0; k < 16; ++k) {
        const int koff = 4 * k + 2 * hi;
        // A fragment: lanes 0-15 rows, v0/v1 = K{koff,koff+1}
        v2f aS = *(const v2f*)(X + (long)row * FEATS + koff);
        v2f aN;
        {
            v2f g = *(const v2f*)(AGG + (long)row * FEATS + koff);
            aN.x = g.x * inv;
            aN.y = g.y * inv;
        }
        #pragma unroll
        for (int n = 0; n < 4; ++n) {
            const int col = n * 16 + r;
            v2f bS, bN;
            bS.x = sWs[koff * FEATS + col];
            bS.y = sWs[(koff + 1) * FEATS + col];
            bN.x = sWn[koff * FEATS + col];
            bN.y = sWn[(koff + 1) * FEATS + col];
            acc[n] = __builtin_amdgcn_wmma_f32_16x16x4_f32(
                false, aS, false, bS, (short)0, acc[n], false, false);
            acc[n] = __builtin_amdgcn_wmma_f32_16x16x4_f32(
                false, aN, false, bN, (short)0, acc[n], false, false);
        }
    }

    // C/D layout: lane L, vgpr vr -> M = vr + 8*(L>>4), N = L&15
    #pragma unroll
    for (int n = 0; n < 4; ++n) {
        const int col = n * 16 + r;
        const float bb = sB[col];
        #pragma unroll
        for (int vr = 0; vr < 8; ++vr) {
            const int m = vr + 8 * hi;
            float v = acc[n][vr] + bb;
            if (do_relu) v = fmaxf(v, 0.0f);
            OUT[(long)(row0 + m) * FEATS + col] = v;
        }
    }
}

// ---------------------------------------------------------------------------
// Final layer (64 -> 1): one wave per node, wave-reduce dot products.
// ---------------------------------------------------------------------------
__global__ void __launch_bounds__(256)
sage_final_kernel(const float* __restrict__ H,
                  const float* __restrict__ AGG,
                  const float* __restrict__ INV,
                  const float* __restrict__ Ws,
                  const float* __restrict__ Wn,
                  const float* __restrict__ b3,
                  float* __restrict__ out, int n) {
    long t = (long)blockIdx.x * blockDim.x + threadIdx.x;
    int node = (int)(t >> 5);
    int lane = (int)(t & 31);
    if (node >= n) return;
    const float inv = INV[node];
    float acc = 0.0f;
    #pragma unroll
    for (int f = lane; f < FEATS; f += 32)
        acc += H[(long)node * FEATS + f] * Ws[f] +
               AGG[(long)node * FEATS + f] * inv * Wn[f];
    #pragma unroll
    for (int off = 16; off > 0; off >>= 1)
        acc += __shfl_down(acc, off, 32);
    if (lane == 0) out[node] = acc + b3[0];
}

// ---------------------------------------------------------------------------
extern "C" void kernel_launch(void* const* d_in, const int* in_sizes, int n_in,
                              void* d_out, int out_size, void* d_ws, size_t ws_size,
                              hipStream_t stream) {
    const float* x    = (const float*)d_in[0];
    const int*   src  = (const int*)d_in[1];
    const int*   dst  = (const int*)d_in[2];
    const float* Ws1  = (const float*)d_in[3];
    const float* Wn1  = (const float*)d_in[4];
    const float* b1   = (const float*)d_in[5];
    const float* Ws2  = (const float*)d_in[6];
    const float* Wn2  = (const float*)d_in[7];
    const float* b2   = (const float*)d_in[8];
    const float* Ws3  = (const float*)d_in[9];
    const float* Wn3  = (const float*)d_in[10];
    const float* b3   = (const float*)d_in[11];
    float* out = (float*)d_out;

    // Workspace layout (floats): inv[50176] | agg[N*64] | h1[N*64] | h2[N*64]
    float* ws  = (float*)d_ws;
    float* inv = ws;
    float* agg = ws + 50176;
    float* h1  = agg + (size_t)N_NODES * FEATS;
    float* h2  = h1 + (size_t)N_NODES * FEATS;

    const size_t aggBytes = (size_t)N_NODES * FEATS * sizeof(float);

    const int degBlocks     = (N_EDGES + 255) / 256;
    const int invBlocks     = (N_NODES + 255) / 256;
    const int scatterBlocks = (int)(((long)N_EDGES * 16 + 255) / 256);
    const int tiles         = (N_NODES + 15) / 16;
    const int combineBlocks = (tiles + 7) / 8;
    const int finalBlocks   = (int)(((long)N_NODES * 32 + 255) / 256);

    // Degrees (shared by all layers)
    hipMemsetAsync(inv, 0, N_NODES * sizeof(float), stream);
    sage_deg_kernel<<<degBlocks, 256, 0, stream>>>(dst, N_EDGES, inv);
    sage_invdeg_kernel<<<invBlocks, 256, 0, stream>>>(inv, N_NODES);

    // Layer 1
    hipMemsetAsync(agg, 0, aggBytes, stream);
    sage_scatter_kernel<<<scatterBlocks, 256, 0, stream>>>(x, src, dst, N_EDGES, agg);
    sage_combine_kernel<<<combineBlocks, 256, 0, stream>>>(
        x, agg, inv, Ws1, Wn1, b1, h1, N_NODES, 1);

    // Layer 2
    hipMemsetAsync(agg, 0, aggBytes, stream);
    sage_scatter_kernel<<<scatterBlocks, 256, 0, stream>>>(h1, src, dst, N_EDGES, agg);
    sage_combine_kernel<<<combineBlocks, 256, 0, stream>>>(
        h1, agg, inv, Ws2, Wn2, b2, h2, N_NODES, 1);

    // Layer 3 (64 -> 1)
    hipMemsetAsync(agg, 0, aggBytes, stream);
    sage_scatter_kernel<<<scatterBlocks, 256, 0, stream>>>(h2, src, dst, N_EDGES, agg);
    sage_final_kernel<<<finalBlocks, 256, 0, stream>>>(
        h2, agg, inv, Ws3, Wn3, b3, out, N_NODES);
}